// LocalAttention_52458730553931
// MI455X (gfx1250) — compile-verified
//
#include <hip/hip_runtime.h>

typedef __attribute__((ext_vector_type(2))) float v2f;
typedef __attribute__((ext_vector_type(8))) float v8f;

#define N_   4
#define C_   64
#define H_   64
#define W_   64
#define G_   4
#define CG_  16            // channels per group
#define HW_  (H_ * W_)
#define FH_  7
#define FW_  7
#define PADT 3
#define PADL 3
#define ROWS_PER_BLK 4
#define TILE_H (ROWS_PER_BLK + FH_ - 1)   // 10

// ---------------------------------------------------------------------------
// CDNA5 async copy: GLOBAL_LOAD_ASYNC_TO_LDS_B32 (tracked by ASYNCcnt).
// vdst = VGPR holding the LDS byte offset; vaddr = 64-bit global address.
// ---------------------------------------------------------------------------
__device__ __forceinline__ void async_g2l_b32(const float* gptr, float* lptr) {
    const unsigned loff =
        (unsigned)reinterpret_cast<unsigned long long>(lptr);   // low 32b = LDS offset
    const unsigned long long ga = reinterpret_cast<unsigned long long>(gptr);
    asm volatile("global_load_async_to_lds_b32 %0, %1, off"
                 :: "v"(loff), "v"(ga)
                 : "memory");
}

__device__ __forceinline__ void wait_asynccnt0() {
    asm volatile("s_wait_asynccnt 0x0" ::: "memory");
}

// ---------------------------------------------------------------------------
// Phase 1: grouped 1x1 conv (q,k,v) via V_WMMA_F32_16X16X4_F32.
// One wave produces one 16(oc) x 16(pos) tile for one (n, group) for all of
// q, k, v simultaneously (B operand = x tile is loaded once, reused 3x).
// D = A(16x4) * B(4x16) + C, accumulated over 4 K-chunks (K total = 16).
// A layout (ISA 7.12.2): lanes 0-15 hold K={0,1} in v0/v1, lanes 16-31 K={2,3}.
// C/D layout: VGPR r holds M = r (lanes 0-15) / M = r+8 (lanes 16-31), N=lane&15.
// ---------------------------------------------------------------------------
__global__ __launch_bounds__(256) void qkv_wmma_kernel(
    const float* __restrict__ x,
    const float* __restrict__ wq,
    const float* __restrict__ wk,
    const float* __restrict__ wv,
    float* __restrict__ q,
    float* __restrict__ k,
    float* __restrict__ v)
{
    const int lane = threadIdx.x & 31;
    const int wave = blockIdx.x * (blockDim.x >> 5) + (threadIdx.x >> 5);

    const int tiles = HW_ / 16;                 // 256 position-tiles per (n,g)
    const int t  = wave % tiles;
    const int g  = (wave / tiles) % G_;
    const int n  = wave / (tiles * G_);
    const int pos0 = t * 16;

    const int half = lane >> 4;                 // 0: lanes 0-15, 1: lanes 16-31
    const int l16  = lane & 15;

    const float* xg = x + ((size_t)n * C_ + (size_t)g * CG_) * HW_ + pos0;
    const int ocb = g * CG_;                    // first global out-channel of group

    v8f accq = {};
    v8f acck = {};
    v8f accv = {};

#pragma unroll
    for (int kc = 0; kc < 4; ++kc) {
        // B chunk: B[kr][nn] = x[n, g*16 + kc*4 + kr, pos0 + nn]
        v2f b;
        b.x = xg[(size_t)(kc * 4 + 0 + 2 * half) * HW_ + l16];
        b.y = xg[(size_t)(kc * 4 + 1 + 2 * half) * HW_ + l16];

        // A chunks: A[oc][kr] = w[(ocb + oc)*CG_ + kc*4 + kr]
        const int wbase = (ocb + l16) * CG_ + kc * 4 + 2 * half;
        v2f aq, ak, av;
        aq.x = wq[wbase]; aq.y = wq[wbase + 1];
        ak.x = wk[wbase]; ak.y = wk[wbase + 1];
        av.x = wv[wbase]; av.y = wv[wbase + 1];

        accq = __builtin_amdgcn_wmma_f32_16x16x4_f32(
            false, aq, false, b, (short)0, accq, false, false);
        acck = __builtin_amdgcn_wmma_f32_16x16x4_f32(
            false, ak, false, b, (short)0, acck, false, false);
        accv = __builtin_amdgcn_wmma_f32_16x16x4_f32(
            false, av, false, b, (short)0, accv, false, false);
    }

    // Store D: element (M = r + 8*half, N = l16)
#pragma unroll
    for (int r = 0; r < 8; ++r) {
        const int oc = r + 8 * half;
        const size_t o = ((size_t)n * C_ + (size_t)(ocb + oc)) * HW_ + pos0 + l16;
        q[o] = accq[r];
        k[o] = acck[r];
        v[o] = accv[r];
    }
}

// ---------------------------------------------------------------------------
// Phase 2: per-channel 7x7 local attention, two-pass softmax (one exp per
// neighbor). One block = (n, c, 4 consecutive rows); 256 threads = 4x64.
// k/v tiles staged in LDS via GLOBAL_LOAD_ASYNC_TO_LDS_B32; zero-padded
// borders reproduce the reference's unfold() semantics (OOB: score = q*r,
// value contribution 0).
// ---------------------------------------------------------------------------
__global__ __launch_bounds__(256) void attn_kernel(
    const float* __restrict__ q,
    const float* __restrict__ k,
    const float* __restrict__ v,
    const float* __restrict__ row_w,   // (49, 32)
    const float* __restrict__ col_w,   // (49, 32)
    float* __restrict__ out)
{
    __shared__ float sk[TILE_H][W_];
    __shared__ float sv[TILE_H][W_];

    const int tid = threadIdx.x;
    const int w   = tid & (W_ - 1);
    const int hl  = tid >> 6;                  // 0..3 row-within-block

    int b = blockIdx.x;
    const int h0 = (b % (H_ / ROWS_PER_BLK)) * ROWS_PER_BLK; b /= (H_ / ROWS_PER_BLK);
    const int c  = b % C_;                     b /= C_;
    const int n  = b;
    const int h  = h0 + hl;

    const size_t base = ((size_t)n * C_ + c) * HW_;

    // Stage k/v tile rows [h0-3, h0+6] into LDS (zero-fill OOB rows).
    for (int idx = tid; idx < TILE_H * W_; idx += 256) {
        const int i  = idx >> 6;               // tile row
        const int ww = idx & (W_ - 1);
        const int gr = h0 - PADT + i;
        if (gr >= 0 && gr < H_) {
            const size_t go = base + (size_t)gr * W_ + ww;
            async_g2l_b32(k + go, &sk[i][ww]);
            async_g2l_b32(v + go, &sv[i][ww]);
        } else {
            sk[i][ww] = 0.0f;
            sv[i][ww] = 0.0f;
        }
    }
    wait_asynccnt0();          // my async LDS writes are done
    __syncthreads();           // (compiler also waits DScnt for the zero-fills)

    // Per-channel positional bias r[c] = sum_p off[p] * {row_w|col_w}[p][c'].
    float rr = 0.0f;
    if (c < 32) {
#pragma unroll
        for (int p = 0; p < FH_ * FW_; ++p)
            rr += (float)(p / FW_ - PADT) * row_w[p * 32 + c];
    } else {
#pragma unroll
        for (int p = 0; p < FH_ * FW_; ++p)
            rr += (float)(p % FW_ - PADL) * col_w[p * 32 + (c - 32)];
    }

    const float qv = q[base + (size_t)h * W_ + w];

    // Pass 1: running max of s = qv*(kp + rr)   (OOB -> kp = 0).
    float m = -__builtin_inff();
#pragma unroll
    for (int dy = 0; dy < FH_; ++dy) {
        const int row = hl + dy;
#pragma unroll
        for (int dx = 0; dx < FW_; ++dx) {
            const int gw  = w - PADL + dx;
            const bool ok = (gw >= 0) && (gw < W_);
            const int gwc = ok ? gw : 0;
            const float kp = ok ? sk[row][gwc] : 0.0f;
            m = fmaxf(m, qv * (kp + rr));
        }
    }

    // Pass 2: single exp per neighbor; accumulate normalizer + weighted V.
    float l = 0.0f;
    float acc = 0.0f;
#pragma unroll
    for (int dy = 0; dy < FH_; ++dy) {
        const int row = hl + dy;
#pragma unroll
        for (int dx = 0; dx < FW_; ++dx) {
            const int gw  = w - PADL + dx;
            const bool ok = (gw >= 0) && (gw < W_);
            const int gwc = ok ? gw : 0;
            const float kp = ok ? sk[row][gwc] : 0.0f;
            const float vp = ok ? sv[row][gwc] : 0.0f;
            const float e  = __expf(qv * (kp + rr) - m);
            l   += e;
            acc += e * vp;
        }
    }

    out[base + (size_t)h * W_ + w] = acc / l;
}

extern "C" void kernel_launch(void* const* d_in, const int* in_sizes, int n_in,
                              void* d_out, int out_size, void* d_ws, size_t ws_size,
                              hipStream_t stream)
{
    const float* x     = (const float*)d_in[0];
    const float* wq    = (const float*)d_in[1];
    const float* wk    = (const float*)d_in[2];
    const float* wv    = (const float*)d_in[3];
    const float* row_w = (const float*)d_in[4];
    const float* col_w = (const float*)d_in[5];
    float* out = (float*)d_out;

    const size_t plane = (size_t)N_ * C_ * HW_;   // 1,048,576 floats
    float* q = (float*)d_ws;
    float* k = q + plane;
    float* v = k + plane;

    // Phase 1: 4096 waves (one 16x16 tile each), 8 waves per 256-thread block.
    const int waves = N_ * G_ * (HW_ / 16);
    qkv_wmma_kernel<<<waves / 8, 256, 0, stream>>>(x, wq, wk, wv, q, k, v);

    // Phase 2: one block per (n, c, 4-row strip).
    attn_kernel<<<N_ * C_ * (H_ / ROWS_PER_BLK), 256, 0, stream>>>(
        q, k, v, row_w, col_w, out);
}